// NTXentLoss_60464549593499
// MI455X (gfx1250) — compile-verified
//
#include <hip/hip_runtime.h>
#include <hip/hip_bf16.h>
#include <math.h>

// NT-Xent loss, fused:
//   K1: row L2-normalize (z1|z2 -> zn)
//   K2: fp32 WMMA gram tiles + mask + online logsumexp, column-chunked for occupancy
//   K3: merge chunk partials -> per-row NLL -> deterministic mean
// N = 4096/view, 2N = 8192 rows, D = 256, fp32 throughout (matches reference).

typedef __attribute__((ext_vector_type(2))) float v2f;
typedef __attribute__((ext_vector_type(8))) float v8f;

#define NROWS   8192       // 2N
#define HALFN   4096
#define DDIM    256
#define RPW     128        // rows per workgroup (8 waves * 16)
#define NT      32         // column-tile width
#define NCHUNK  4          // column chunks (grid.y) -> 256 WGs total
#define TILES_PER_CHUNK ((NROWS / NT) / NCHUNK)   // 64
#define LDSTR   260        // LDS row stride floats (256 + 4 pad: 16B-aligned, bank-clean)
#define TEMP_INV 2.0f      // 1 / 0.5
#define MASKED  (-1.8e16f) // -9e15 / 0.5

// ---------------- Kernel 1: row L2-normalize, concat z1|z2 -> zn ----------------
__global__ __launch_bounds__(256) void ntxent_normalize(const float* __restrict__ z1,
                                                        const float* __restrict__ z2,
                                                        float* __restrict__ zn) {
  const int wid  = (blockIdx.x * 256 + threadIdx.x) >> 5;   // one wave per row
  const int lane = threadIdx.x & 31;
  const float* src = (wid < HALFN) ? (z1 + (size_t)wid * DDIM)
                                   : (z2 + (size_t)(wid - HALFN) * DDIM);
  float v[8];
  float ss = 0.f;
#pragma unroll
  for (int i = 0; i < 8; ++i) { v[i] = src[lane + i * 32]; ss += v[i] * v[i]; }
#pragma unroll
  for (int off = 1; off < 32; off <<= 1) ss += __shfl_xor(ss, off, 32);
  const float inv = 1.0f / fmaxf(sqrtf(ss), 1e-8f);
  float* dst = zn + (size_t)wid * DDIM;
#pragma unroll
  for (int i = 0; i < 8; ++i) dst[lane + i * 32] = v[i] * inv;
}

// ------- Kernel 2: fused zn@zn^T (fp32 WMMA) + mask + online logsumexp + pos -------
// grid = (64, NCHUNK): blockIdx.x picks the 128-row panel, blockIdx.y a 2048-col chunk.
// Writes per-(row,chunk) partials: pm (running max), ps (sumexp), pp (positive logit).
__global__ __launch_bounds__(256) void ntxent_sim_lse(const float* __restrict__ zn,
                                                      float* __restrict__ pm,
                                                      float* __restrict__ ps,
                                                      float* __restrict__ pp) {
  __shared__ float Bs[NT * LDSTR];                 // 33,280 B

  const int tid  = threadIdx.x;
  const int wave = tid >> 5;
  const int lane = tid & 31;
  const int l15  = lane & 15;
  const int hi   = lane >> 4;                      // 0: K0..1 / M0..7   1: K2..3 / M8..15
  const int rowBase = blockIdx.x * RPW + wave * 16;
  const int chunk   = blockIdx.y;

  // --- A panel (16 rows x 256) resident in registers: 64 v2f frags ---
  v2f afrag[DDIM / 4];
  {
    const float* arow = zn + (size_t)(rowBase + l15) * DDIM + hi * 2;
#pragma unroll
    for (int j = 0; j < DDIM / 4; ++j)
      afrag[j] = *(const v2f*)(arow + 4 * j);
  }

  float m[8], s[8], pos[8];
#pragma unroll
  for (int r = 0; r < 8; ++r) { m[r] = -INFINITY; s[r] = 0.f; pos[r] = 0.f; }

  const int ct0 = chunk * TILES_PER_CHUNK;
  for (int t = 0; t < TILES_PER_CHUNK; ++t) {
    const int c0 = (ct0 + t) * NT;

    __syncthreads();
    { // cooperative B-tile stage: 32 rows of zn -> LDS (8 threads/row, float4)
      const int r  = tid >> 3;
      const int cq = (tid & 7) * 8;                // float4 index base
      const float4* src = (const float4*)(zn + (size_t)(c0 + r) * DDIM) + cq;
      float4* dst = (float4*)(&Bs[r * LDSTR]) + cq;
#pragma unroll
      for (int q = 0; q < 8; ++q) dst[q] = src[q];
    }
    __syncthreads();

    // 4 independent accumulator chains (2 col-subtiles x even/odd k-steps)
    v8f a0e = {}, a0o = {}, a1e = {}, a1o = {};
    const float* b0 = &Bs[(0 * 16 + l15) * LDSTR + hi * 2];
    const float* b1 = &Bs[(1 * 16 + l15) * LDSTR + hi * 2];
#pragma unroll
    for (int j = 0; j < DDIM / 4; j += 2) {        // 32 double k-steps
      v2f bf0e = *(const v2f*)(b0 + 4 * j);
      v2f bf0o = *(const v2f*)(b0 + 4 * (j + 1));
      v2f bf1e = *(const v2f*)(b1 + 4 * j);
      v2f bf1o = *(const v2f*)(b1 + 4 * (j + 1));
      a0e = __builtin_amdgcn_wmma_f32_16x16x4_f32(false, afrag[j],     false, bf0e,
                                                  (short)0, a0e, false, false);
      a1e = __builtin_amdgcn_wmma_f32_16x16x4_f32(false, afrag[j],     false, bf1e,
                                                  (short)0, a1e, false, false);
      a0o = __builtin_amdgcn_wmma_f32_16x16x4_f32(false, afrag[j + 1], false, bf0o,
                                                  (short)0, a0o, false, false);
      a1o = __builtin_amdgcn_wmma_f32_16x16x4_f32(false, afrag[j + 1], false, bf1o,
                                                  (short)0, a1o, false, false);
    }
    const v8f acc0 = a0e + a0o;
    const v8f acc1 = a1e + a1o;

    // epilogue: temperature, diagonal mask, positive capture, online (m,s)
#pragma unroll
    for (int cs = 0; cs < 2; ++cs) {
      const v8f c = cs ? acc1 : acc0;
      const int col_g = c0 + cs * 16 + l15;
#pragma unroll
      for (int r = 0; r < 8; ++r) {
        const int row_g = rowBase + r + hi * 8;
        float val = c[r] * TEMP_INV;
        if (row_g == col_g) val = MASKED;
        if (col_g == ((row_g + HALFN) & (NROWS - 1))) pos[r] = val;
        const float mo = m[r];
        const float mn = fmaxf(mo, val);
        s[r] = s[r] * __expf(mo - mn) + __expf(val - mn);
        m[r] = mn;
      }
    }
  }

  // merge the 16 lanes of each half-wave (butterfly), write per-(row,chunk) partials
#pragma unroll
  for (int r = 0; r < 8; ++r) {
    float mv = m[r], sv = s[r], pv = pos[r];
#pragma unroll
    for (int off = 1; off < 16; off <<= 1) {
      const float mo = __shfl_xor(mv, off, 32);
      const float so = __shfl_xor(sv, off, 32);
      const float po = __shfl_xor(pv, off, 32);
      const float mn = fmaxf(mv, mo);
      sv = sv * __expf(mv - mn) + so * __expf(mo - mn);
      mv = mn;
      pv += po;
    }
    if (l15 == 0) {
      const int row_g = rowBase + r + hi * 8;
      const size_t idx = (size_t)chunk * NROWS + row_g;
      pm[idx] = mv;
      ps[idx] = sv;
      pp[idx] = pv;
    }
  }
}

// ------- Kernel 3: merge chunk partials -> per-row NLL -> deterministic mean -------
__global__ __launch_bounds__(256) void ntxent_finish(const float* __restrict__ pm,
                                                     const float* __restrict__ ps,
                                                     const float* __restrict__ pp,
                                                     float* __restrict__ out) {
  __shared__ float sm[256];
  float acc = 0.f;
  for (int row = threadIdx.x; row < NROWS; row += 256) {
    float mv = -INFINITY, sv = 0.f, pv = 0.f;
#pragma unroll
    for (int c = 0; c < NCHUNK; ++c) {
      const size_t idx = (size_t)c * NROWS + row;
      const float mc = pm[idx];
      const float sc = ps[idx];
      const float mn = fmaxf(mv, mc);
      sv = sv * __expf(mv - mn) + sc * __expf(mc - mn);
      mv = mn;
      pv += pp[idx];
    }
    acc += (-pv + mv + logf(sv));
  }
  sm[threadIdx.x] = acc;
  __syncthreads();
  for (int st = 128; st > 0; st >>= 1) {
    if (threadIdx.x < st) sm[threadIdx.x] += sm[threadIdx.x + st];
    __syncthreads();
  }
  if (threadIdx.x == 0) out[0] = sm[0] * (1.0f / NROWS);
}

extern "C" void kernel_launch(void* const* d_in, const int* in_sizes, int n_in,
                              void* d_out, int out_size, void* d_ws, size_t ws_size,
                              hipStream_t stream) {
  const float* z1 = (const float*)d_in[0];
  const float* z2 = (const float*)d_in[1];

  float* zn = (float*)d_ws;                                   // 8 MB
  char*  p  = (char*)d_ws + (size_t)NROWS * DDIM * sizeof(float);
  float* pm = (float*)p;                                      // 4*8192*4 = 128 KB
  float* ps = (float*)(p + (size_t)NCHUNK * NROWS * sizeof(float));
  float* pp = (float*)(p + 2 * (size_t)NCHUNK * NROWS * sizeof(float));
  float* out = (float*)d_out;

  ntxent_normalize<<<NROWS / 8, 256, 0, stream>>>(z1, z2, zn);      // 1024 blocks
  dim3 grid2(NROWS / RPW, NCHUNK);                                  // 64 x 4 = 256 WGs
  ntxent_sim_lse<<<grid2, 256, 0, stream>>>(zn, pm, ps, pp);
  ntxent_finish<<<1, 256, 0, stream>>>(pm, ps, pp, out);
}